// DropBlockT_1d_48352741818430
// MI455X (gfx1250) — compile-verified
//
#include <hip/hip_runtime.h>
#include <hip/hip_bf16.h>

// DropBlockT_1d forward for MI455X (gfx1250).
// Stage 1 (tiny): reproduce JAX threefry bernoulli + maxpool + rescale into a
//                 per-(n,t) scale table kscale[9600] in d_ws.
// Stage 2 (hot):  out = in * kscale[n,t], pure HBM-bound streaming with
//                 b128 non-temporal loads/stores, global_prefetch, and
//                 async global->LDS staging of the scale table (ASYNCcnt).

#define N_  32
#define C_  256
#define T_  300
#define V_  25
#define NT_      (N_ * T_)   // 9600
#define HALF_NT_ (NT_ / 2)   // 4800
#define PLANE_   (T_ * V_)   // 7500 floats, 16B aligned per plane
#define NF4_     (PLANE_ / 4) // 1875 float4 per plane

typedef float f4 __attribute__((ext_vector_type(4)));
typedef int   v4i __attribute__((vector_size(16)));   // matches builtin param

#if __has_builtin(__builtin_amdgcn_global_load_async_to_lds_b128) && \
    __has_builtin(__builtin_amdgcn_s_wait_asynccnt)
#define HAS_ASYNC_LDS 1
#else
#define HAS_ASYNC_LDS 0
#endif

__device__ __forceinline__ unsigned rotl32(unsigned x, unsigned r) {
  return (x << r) | (x >> (32u - r));
}

// Exact JAX threefry2x32 (20 rounds). key(42) -> (k0,k1) = (0,42).
__device__ __forceinline__ void threefry2x32(unsigned k0, unsigned k1,
                                             unsigned x0, unsigned x1,
                                             unsigned& o0, unsigned& o1) {
  const unsigned ks2 = k0 ^ k1 ^ 0x1BD11BDAu;
  unsigned y0 = x0 + k0, y1 = x1 + k1;
#define TF_R(r) { y0 += y1; y1 = rotl32(y1, r); y1 ^= y0; }
  TF_R(13) TF_R(15) TF_R(26) TF_R(6)
  y0 += k1;  y1 += ks2 + 1u;
  TF_R(17) TF_R(29) TF_R(16) TF_R(24)
  y0 += ks2; y1 += k0 + 2u;
  TF_R(13) TF_R(15) TF_R(26) TF_R(6)
  y0 += k0;  y1 += k1 + 3u;
  TF_R(17) TF_R(29) TF_R(16) TF_R(24)
  y0 += k1;  y1 += ks2 + 4u;
  TF_R(13) TF_R(15) TF_R(26) TF_R(6)
  y0 += ks2; y1 += k0 + 5u;
#undef TF_R
  o0 = y0; o1 = y1;
}

// ---------------------------------------------------------------------------
// Kernel 1: one block, 1024 threads (32 waves). Builds kscale[NT_] in d_ws.
// ---------------------------------------------------------------------------
__global__ __launch_bounds__(1024)
void dropblock_mask_kernel(const float* __restrict__ mask,
                           float* __restrict__ kscale) {
  __shared__ float sM[NT_];     // 38.4 KB bernoulli seeds
  __shared__ float red[1024];   // reduction scratch
  const int tid = threadIdx.x;

  // --- sum(mask) ---
  float s = 0.f;
  for (int i = tid; i < NT_; i += 1024) s += mask[i];
  red[tid] = s;
  __syncthreads();
  for (int off = 512; off > 0; off >>= 1) {
    if (tid < off) red[tid] += red[tid + off];
    __syncthreads();
  }
  const float sumMask = red[0];
  __syncthreads();  // everyone read red[0] before reuse

  // p = min(1, mask * (NT_/sumMask) * gamma), gamma = (1-0.9)/7
  const float pfac = ((float)NT_ / sumMask) * (0.1f / 7.0f);

  // --- bernoulli seeds: bits[i] pairs with bits[i+4800] (JAX iota split) ---
  for (int i = tid; i < HALF_NT_; i += 1024) {
    unsigned b0, b1;
    threefry2x32(0u, 42u, (unsigned)i, (unsigned)(i + HALF_NT_), b0, b1);
    float u0 = __uint_as_float((b0 >> 9) | 0x3f800000u) - 1.0f;
    float u1 = __uint_as_float((b1 >> 9) | 0x3f800000u) - 1.0f;
    u0 = fmaxf(u0, 0.0f);
    u1 = fmaxf(u1, 0.0f);
    const float p0 = fminf(mask[i] * pfac, 1.0f);
    const float p1 = fminf(mask[i + HALF_NT_] * pfac, 1.0f);
    sM[i]            = (u0 < p0) ? 1.0f : 0.0f;
    sM[i + HALF_NT_] = (u1 < p1) ? 1.0f : 0.0f;
  }
  __syncthreads();

  // --- maxpool1d(k=7, pad=3) over t; keep = 1 - pool; reduce sum(keep) ---
  float keepv[10];  // ceil(9600/1024) = 10 per thread
  float sk = 0.f;
  int cnt = 0;
  for (int j = tid; j < NT_; j += 1024, ++cnt) {
    const int nn = j / T_;
    const int t  = j - nn * T_;
    int lo = t - 3; if (lo < 0) lo = 0;
    int hi = t + 3; if (hi > T_ - 1) hi = T_ - 1;
    float mx = 0.f;  // zero padding == -inf padding since M >= 0
    const float* row = &sM[nn * T_];
    for (int tt = lo; tt <= hi; ++tt) mx = fmaxf(mx, row[tt]);
    const float kp = 1.0f - mx;
    keepv[cnt] = kp;
    sk += kp;
  }
  red[tid] = sk;
  __syncthreads();
  for (int off = 512; off > 0; off >>= 1) {
    if (tid < off) red[tid] += red[tid + off];
    __syncthreads();
  }
  // scale = (n*c*v*t) / (sum(keep) * c*v) = NT_ / sum(keep)
  const float scale = (float)NT_ / red[0];

  cnt = 0;
  for (int j = tid; j < NT_; j += 1024, ++cnt)
    kscale[j] = keepv[cnt] * scale;
}

// ---------------------------------------------------------------------------
// Kernel 2: one block per (n,c) plane of 7500 contiguous floats.
// out = in * kscale[n,t], HBM-bound streaming.
// ---------------------------------------------------------------------------
__global__ __launch_bounds__(256)
void dropblock_apply_kernel(const float* __restrict__ in,
                            const float* __restrict__ kscale,
                            float* __restrict__ out) {
  __shared__ __align__(16) float ksh[T_];  // this n's 300 scale factors
  const int tid   = threadIdx.x;
  const int plane = blockIdx.x;     // n*C_ + c
  const int n     = plane >> 8;     // C_ == 256

  // Stage kscale row into LDS via CDNA5 async global->LDS (ASYNCcnt path).
#if HAS_ASYNC_LDS
  if (tid < T_ / 4) {  // 75 lanes x b128 = 300 floats
    v4i* gsrc = (v4i*)(kscale + n * T_) + tid;
    v4i* ldst = (v4i*)&ksh[tid * 4];
    __builtin_amdgcn_global_load_async_to_lds_b128(gsrc, ldst,
                                                   /*offset=*/0, /*cpol=*/0);
  }
  __builtin_amdgcn_s_wait_asynccnt(0);
  __syncthreads();
#else
  for (int i = tid; i < T_; i += 256) ksh[i] = kscale[n * T_ + i];
  __syncthreads();
#endif

  const long long base = (long long)plane * PLANE_;
  const f4* __restrict__ in4 = (const f4*)(in + base);
  f4* __restrict__ out4      = (f4*)(out + base);

  for (int i = tid; i < NF4_; i += 256) {
    if (i + 256 < NF4_) __builtin_prefetch(&in4[i + 256], 0, 0);
    f4 x = __builtin_nontemporal_load(&in4[i]);
    // a float4 crosses at most one t boundary (V_ = 25)
    const int l0  = i * 4;
    const int t0  = l0 / 25;
    const int rem = l0 - t0 * 25;          // 0..24
    const float k0 = ksh[t0];
    const float kn = (rem >= 22) ? ksh[t0 + 1] : k0;  // only fetched if needed
    f4 y;
    y.x = x.x * k0;
    y.y = x.y * ((rem >= 24) ? kn : k0);
    y.z = x.z * ((rem >= 23) ? kn : k0);
    y.w = x.w * ((rem >= 22) ? kn : k0);
    __builtin_nontemporal_store(y, &out4[i]);
  }
}

// ---------------------------------------------------------------------------
extern "C" void kernel_launch(void* const* d_in, const int* in_sizes, int n_in,
                              void* d_out, int out_size, void* d_ws, size_t ws_size,
                              hipStream_t stream) {
  const float* input = (const float*)d_in[0];  // (32,256,300,25) f32
  const float* mask  = (const float*)d_in[1];  // (32,1,300)      f32
  float* out    = (float*)d_out;
  float* kscale = (float*)d_ws;                // 9600 floats of scratch

  hipLaunchKernelGGL(dropblock_mask_kernel, dim3(1), dim3(1024), 0, stream,
                     mask, kscale);
  hipLaunchKernelGGL(dropblock_apply_kernel, dim3(N_ * C_), dim3(256), 0, stream,
                     input, kscale, out);
}